// LengthRegulator_2336462209235
// MI455X (gfx1250) — compile-verified
//
#include <hip/hip_runtime.h>
#include <hip/hip_bf16.h>

typedef __attribute__((ext_vector_type(2))) float v2f;
typedef __attribute__((ext_vector_type(4))) float v4f;
typedef __attribute__((ext_vector_type(8))) float v8f;
typedef int v4i_ __attribute__((vector_size(16)));   // matches async builtin param

#define B_ 64
#define L_ 512
#define D_ 256
#define F_ 256
#define T_ 4096
#define K_ 768
#define M_ (B_ * L_)     // 32768
#define BLK_M 64
#define KC 32
#define NSTAGE (K_ / KC) // 24

// ---- CDNA5 async global->LDS path (guarded; falls back to sync copies) ----
#if __has_builtin(__builtin_amdgcn_global_load_async_to_lds_b128)
#define HAVE_ASYNC_LDS 1
__device__ __forceinline__ void async_cp16(const float4* g, float4* l) {
  __builtin_amdgcn_global_load_async_to_lds_b128(
      (__attribute__((address_space(1))) v4i_*)g,
      (__attribute__((address_space(3))) v4i_*)l, 0, 0);
}
__device__ __forceinline__ void async_wait0() {
#if __has_builtin(__builtin_amdgcn_s_wait_asynccnt)
  __builtin_amdgcn_s_wait_asynccnt(0);
#else
  asm volatile("s_wait_asynccnt 0" ::: "memory");
#endif
}
#else
#define HAVE_ASYNC_LDS 0
#endif

// ---------------------------------------------------------------------------
// cumsum of durations per batch row (Hillis-Steele scan in LDS)
// ---------------------------------------------------------------------------
__global__ __launch_bounds__(512) void scan_kernel(const int* __restrict__ target,
                                                   int* __restrict__ ends) {
  __shared__ int s[L_];
  int b = blockIdx.x;
  int tid = threadIdx.x;            // 512 threads
  s[tid] = target[b * L_ + tid];
  __syncthreads();
  for (int off = 1; off < L_; off <<= 1) {
    int v = (tid >= off) ? s[tid - off] : 0;
    __syncthreads();
    s[tid] += v;
    __syncthreads();
  }
  ends[b * L_ + tid] = s[tid];
}

// ---------------------------------------------------------------------------
// length regulate: one wave32 per output frame, binary search in LDS ends,
// non-temporal streaming stores (output >> L2; keep L2 for x/weights)
// ---------------------------------------------------------------------------
__global__ __launch_bounds__(256) void lreg_kernel(const float* __restrict__ x,
                                                   const int* __restrict__ ends,
                                                   float* __restrict__ out) {
  __shared__ int se[L_];
  int tid = threadIdx.x;            // 256 threads = 8 waves
  int row0 = blockIdx.x * 8;        // 8 frames per block, same batch (T%8==0)
  int b = row0 >> 12;               // / T_
  se[tid]       = ends[b * L_ + tid];
  se[tid + 256] = ends[b * L_ + tid + 256];
  __syncthreads();
  int w = tid >> 5, lane = tid & 31;
  int t = (row0 & (T_ - 1)) + w;
  // searchsorted(ends, t, side='right'): first idx with ends[idx] > t
  int lo = 0, hi = L_;
  while (lo < hi) {
    int mid = (lo + hi) >> 1;
    if (se[mid] <= t) lo = mid + 1; else hi = mid;
  }
  int idx = lo < (L_ - 1) ? lo : (L_ - 1);
  bool valid = t < se[L_ - 1];
  const v4f* src = (const v4f*)(x + (size_t)(b * L_ + idx) * D_);
  v4f* dst = (v4f*)(out + ((size_t)b * T_ + t) * D_);
  const v4f z = {0.f, 0.f, 0.f, 0.f};
  v4f v0 = valid ? src[lane]      : z;
  v4f v1 = valid ? src[lane + 32] : z;
  __builtin_nontemporal_store(v0, dst + lane);
  __builtin_nontemporal_store(v1, dst + lane + 32);
}

// ---------------------------------------------------------------------------
// pack conv weights (768x256 row-major) into WMMA-fragment order:
//   Wp[((k>>2)*256 + n)*4 + (k&3)]  -> per k-step, per column, 4 k contiguous
// ---------------------------------------------------------------------------
__global__ __launch_bounds__(256) void pack_w_kernel(const float* __restrict__ W,
                                                     float* __restrict__ Wp) {
  int idx = blockIdx.x * 256 + threadIdx.x;   // over 768*64 float4s
  int k  = idx >> 6;
  int n4 = idx & 63;
  v4f v = ((const v4f*)W)[idx];
  int ks = k >> 2, pos = k & 3;
  float* dst = Wp + ((size_t)(ks * 256 + n4 * 4)) * 4 + pos;
  dst[0]  = v.x;
  dst[4]  = v.y;
  dst[8]  = v.z;
  dst[12] = v.w;
}

// ---------------------------------------------------------------------------
// conv1d(k=3,same) as GEMM via implicit im2col, f32 WMMA 16x16x4.
// A row m=(b,l): 768 contiguous floats at Ain[b, l-1, 0] (edge taps zeroed).
// Double-buffered LDS stages; fragment-native layouts -> ds_load_b64 frags.
// Block: 256 thr / 8 waves, tile 64M x 256N; wave = 16M x 128N (8 wmma tiles)
// ---------------------------------------------------------------------------
__global__ __launch_bounds__(256) void conv_gemm_kernel(const float* __restrict__ Ain,
                                                        const float* __restrict__ Wp,
                                                        const float* __restrict__ bias,
                                                        float* __restrict__ out) {
  __shared__ float Asp[2][8][BLK_M][4];   // 16 KB: [buf][kstep][m][k-in-step]
  __shared__ float Bsp[2][8][F_][4];      // 64 KB: [buf][kstep][n][k-in-step]

  int tid  = threadIdx.x;
  int wv   = tid >> 5;
  int lane = tid & 31;
  int half = lane >> 4;
  int mrow = lane & 15;

  int m_base = blockIdx.x * BLK_M;
  int m_off  = (wv >> 1) * 16;
  int n_off  = (wv & 1) * 128;

  const v8f vzero = {0.f, 0.f, 0.f, 0.f, 0.f, 0.f, 0.f, 0.f};
  v8f acc[8];
#pragma unroll
  for (int i = 0; i < 8; ++i) acc[i] = vzero;

  // A loader mapping: thread -> (row 0..63, 8-float k-chunk); chunks never straddle a tap
  int a_row  = tid >> 2;
  int a_kc   = (tid & 3) * 8;
  int m_glob = m_base + a_row;
  int bb = m_glob >> 9;
  int ll = m_glob & (L_ - 1);
  const float4 z4 = make_float4(0.f, 0.f, 0.f, 0.f);

  auto load_stage = [&](int s, int bi) {
    // ---- A tile (with same-padding zero fill on edge taps) ----
    int kg  = s * KC + a_kc;
    int tap = kg >> 8;
    int c   = kg & 255;
    int ls  = ll + tap - 1;
    bool valid = (ls >= 0) && (ls < L_);
    int lsc = ls < 0 ? 0 : (ls > (L_ - 1) ? (L_ - 1) : ls);
    const float4* ag = (const float4*)(Ain + (size_t)(bb * L_ + lsc) * D_ + c);
    int ks0 = a_kc >> 2;
    float4* ad0 = (float4*)&Asp[bi][ks0][a_row][0];
    float4* ad1 = (float4*)&Asp[bi][ks0 + 1][a_row][0];
    // ---- B tile: straight copy of pre-packed weights (32 KB contiguous) ----
    const float4* bg = (const float4*)(Wp + (size_t)s * KC * F_);
    float4* bd = (float4*)&Bsp[bi][0][0][0];
#if HAVE_ASYNC_LDS
    if (valid) {
      async_cp16(ag, ad0);
      async_cp16(ag + 1, ad1);
    } else {
      *ad0 = z4;
      *ad1 = z4;
    }
#pragma unroll
    for (int j = 0; j < 8; ++j) async_cp16(bg + tid + j * 256, bd + tid + j * 256);
#else
    float4 v0 = ag[0], v1 = ag[1];
    if (!valid) { v0 = z4; v1 = z4; }
    *ad0 = v0;
    *ad1 = v1;
#pragma unroll
    for (int j = 0; j < 8; ++j) bd[tid + j * 256] = bg[tid + j * 256];
#endif
  };

  load_stage(0, 0);

  for (int s = 0; s < NSTAGE; ++s) {
    int bi = s & 1;
#if HAVE_ASYNC_LDS
    async_wait0();
#endif
    __syncthreads();                       // stage s visible; stage s-1 reads done
    if (s + 1 < NSTAGE) load_stage(s + 1, bi ^ 1);

#pragma unroll
    for (int ks = 0; ks < 8; ++ks) {
      v2f af = *(const v2f*)&Asp[bi][ks][m_off + mrow][2 * half];
#pragma unroll
      for (int nt = 0; nt < 8; ++nt) {
        v2f bf = *(const v2f*)&Bsp[bi][ks][n_off + nt * 16 + mrow][2 * half];
        acc[nt] = __builtin_amdgcn_wmma_f32_16x16x4_f32(
            false, af, false, bf, (short)0, acc[nt], false, false);
      }
    }
  }

  // ---- epilogue: add bias, store (C layout: vgpr v -> M=v+8*half, lane%16 -> N) ----
#pragma unroll
  for (int nt = 0; nt < 8; ++nt) {
    int n = n_off + nt * 16 + mrow;
    float bv = bias[n];
#pragma unroll
    for (int v = 0; v < 8; ++v) {
      int m = m_base + m_off + v + 8 * half;
      out[(size_t)m * F_ + n] = acc[nt][v] + bv;
    }
  }
}

// ---------------------------------------------------------------------------
// LayerNorm(channel) + ReLU, one wave32 per row of 256
// ---------------------------------------------------------------------------
__device__ __forceinline__ void ln_row(float4 a, float4 b, int lane,
                                       const float* g, const float* be,
                                       float4& o0, float4& o1) {
  float s = a.x + a.y + a.z + a.w + b.x + b.y + b.z + b.w;
  float q = a.x * a.x + a.y * a.y + a.z * a.z + a.w * a.w +
            b.x * b.x + b.y * b.y + b.z * b.z + b.w * b.w;
#pragma unroll
  for (int off = 16; off > 0; off >>= 1) {
    s += __shfl_xor(s, off, 32);
    q += __shfl_xor(q, off, 32);
  }
  float mu  = s * (1.0f / 256.0f);
  float inv = rsqrtf(q * (1.0f / 256.0f) - mu * mu + 1e-5f);
  float4 g0 = ((const float4*)g)[lane], g1 = ((const float4*)g)[lane + 32];
  float4 b0 = ((const float4*)be)[lane], b1 = ((const float4*)be)[lane + 32];
  o0.x = fmaxf((a.x - mu) * inv * g0.x + b0.x, 0.f);
  o0.y = fmaxf((a.y - mu) * inv * g0.y + b0.y, 0.f);
  o0.z = fmaxf((a.z - mu) * inv * g0.z + b0.z, 0.f);
  o0.w = fmaxf((a.w - mu) * inv * g0.w + b0.w, 0.f);
  o1.x = fmaxf((b.x - mu) * inv * g1.x + b1.x, 0.f);
  o1.y = fmaxf((b.y - mu) * inv * g1.y + b1.y, 0.f);
  o1.z = fmaxf((b.z - mu) * inv * g1.z + b1.z, 0.f);
  o1.w = fmaxf((b.w - mu) * inv * g1.w + b1.w, 0.f);
}

__global__ __launch_bounds__(256) void ln_relu_kernel(const float* __restrict__ tin,
                                                      const float* __restrict__ g,
                                                      const float* __restrict__ be,
                                                      float* __restrict__ hout) {
  int row = blockIdx.x * 8 + (threadIdx.x >> 5);
  int lane = threadIdx.x & 31;
  const float4* src = (const float4*)(tin + (size_t)row * F_);
  float4 a = src[lane], b = src[lane + 32];
  float4 o0, o1;
  ln_row(a, b, lane, g, be, o0, o1);
  float4* dst = (float4*)(hout + (size_t)row * F_);
  dst[lane] = o0;
  dst[lane + 32] = o1;
}

// LayerNorm + ReLU + fused (256 -> 1) linear + ReLU  ->  dp_out[row]
__global__ __launch_bounds__(256) void ln_relu_lin_kernel(const float* __restrict__ tin,
                                                          const float* __restrict__ g,
                                                          const float* __restrict__ be,
                                                          const float* __restrict__ lw,
                                                          const float* __restrict__ lb,
                                                          float* __restrict__ dp) {
  int row = blockIdx.x * 8 + (threadIdx.x >> 5);
  int lane = threadIdx.x & 31;
  const float4* src = (const float4*)(tin + (size_t)row * F_);
  float4 a = src[lane], b = src[lane + 32];
  float4 o0, o1;
  ln_row(a, b, lane, g, be, o0, o1);
  float4 w0 = ((const float4*)lw)[lane], w1 = ((const float4*)lw)[lane + 32];
  float d = o0.x * w0.x + o0.y * w0.y + o0.z * w0.z + o0.w * w0.w +
            o1.x * w1.x + o1.y * w1.y + o1.z * w1.z + o1.w * w1.w;
#pragma unroll
  for (int off = 16; off > 0; off >>= 1) d += __shfl_xor(d, off, 32);
  if (lane == 0) dp[row] = fmaxf(d + lb[0], 0.f);
}

// ---------------------------------------------------------------------------
extern "C" void kernel_launch(void* const* d_in, const int* in_sizes, int n_in,
                              void* d_out, int out_size, void* d_ws, size_t ws_size,
                              hipStream_t stream) {
  const float* x   = (const float*)d_in[0];
  const int*   tgt = (const int*)d_in[1];
  // d_in[2] = mel_max_length (compile-time T_=4096 per reference)
  const float* w1  = (const float*)d_in[3];
  const float* b1  = (const float*)d_in[4];
  const float* g1  = (const float*)d_in[5];
  const float* be1 = (const float*)d_in[6];
  const float* w2  = (const float*)d_in[7];
  const float* b2  = (const float*)d_in[8];
  const float* g2  = (const float*)d_in[9];
  const float* be2 = (const float*)d_in[10];
  const float* lw  = (const float*)d_in[11];
  const float* lb  = (const float*)d_in[12];

  float* out = (float*)d_out;                       // (B, T, D)
  float* dp  = out + (size_t)B_ * T_ * D_;          // (B, L)

  float* t1   = (float*)d_ws;                       // (M, 256) GEMM scratch
  float* h    = t1 + (size_t)M_ * F_;               // (M, 256) hidden
  int*   ends = (int*)(h + (size_t)M_ * F_);        // (B, L) cumsum
  float* Wp   = (float*)(ends + (size_t)B_ * L_);   // (768, 256) packed weights

  // length-regulator path (pure streaming, memory-bound)
  scan_kernel<<<B_, L_, 0, stream>>>(tgt, ends);
  lreg_kernel<<<(B_ * T_) / 8, 256, 0, stream>>>(x, ends, out);

  // duration-predictor path (f32 WMMA GEMMs + fused LN epilogues)
  pack_w_kernel<<<(K_ * F_ / 4) / 256, 256, 0, stream>>>(w1, Wp);
  conv_gemm_kernel<<<M_ / BLK_M, 256, 0, stream>>>(x, Wp, b1, t1);
  ln_relu_kernel<<<M_ / 8, 256, 0, stream>>>(t1, g1, be1, h);
  pack_w_kernel<<<(K_ * F_ / 4) / 256, 256, 0, stream>>>(w2, Wp);
  conv_gemm_kernel<<<M_ / BLK_M, 256, 0, stream>>>(h, Wp, b2, t1);
  ln_relu_lin_kernel<<<M_ / 8, 256, 0, stream>>>(t1, g2, be2, lw, lb, dp);
}